// Policy_Network_56495999812263
// MI455X (gfx1250) — compile-verified
//
#include <hip/hip_runtime.h>
#include <math.h>

typedef __attribute__((ext_vector_type(2))) float v2f;
typedef __attribute__((ext_vector_type(4))) float v4f;
typedef __attribute__((ext_vector_type(8))) float v8f;

#define B_  64
#define S_  32
#define R_  128
#define A_  64
#define DW_ 512
#define DR_ 512

__device__ __forceinline__ float fast_tanh(float x) {
#if __has_builtin(__builtin_amdgcn_tanhf)
  return __builtin_amdgcn_tanhf(x);
#else
  return tanhf(x);
#endif
}

// ---------------------------------------------------------------------------
// NT GEMM with V_WMMA_F32_16X16X4_F32:
//   out[m][n] = sum_k A[m][k] * Bm[n][k] + bias[n]  (+ cov[m]*wc[n] + bc[n])
// grid.x = M/64 (4 waves per block, one 16-row tile per wave), grid.y = N/16.
// A-frag layout (16x4 f32): lanes 0-15 hold M=0..15 {K0,K1}; lanes 16-31 {K2,K3}.
// B-frag (4x16) is the mirror image, so NT addressing is identical for A and B.
// ---------------------------------------------------------------------------
__global__ void gemm_nt_wmma(const float* __restrict__ A,
                             const float* __restrict__ Bm,
                             float* __restrict__ out,
                             int M, int N, int K,
                             const float* __restrict__ bias,
                             const float* __restrict__ cov,
                             const float* __restrict__ wc,
                             const float* __restrict__ bcv) {
  const int t      = threadIdx.x;
  const int lane   = t & 31;
  const int lo     = lane & 15;
  const int hi     = lane >> 4;
  const int tile_m = blockIdx.x * 4 + (t >> 5);
  const int tile_n = blockIdx.y;

  const float* Arow = A  + (size_t)(tile_m * 16 + lo) * K;
  const float* Brow = Bm + (size_t)(tile_n * 16 + lo) * K;

  v8f c = {};
  for (int k = 0; k < K; k += 8) {
    v4f av = *(const v4f*)(Arow + k + 4 * hi);
    v4f bv = *(const v4f*)(Brow + k + 4 * hi);
    v2f a0 = __builtin_shufflevector(av, av, 0, 1);
    v2f b0 = __builtin_shufflevector(bv, bv, 0, 1);
    v2f a1 = __builtin_shufflevector(av, av, 2, 3);
    v2f b1 = __builtin_shufflevector(bv, bv, 2, 3);
    c = __builtin_amdgcn_wmma_f32_16x16x4_f32(false, a0, false, b0, (short)0, c, false, false);
    c = __builtin_amdgcn_wmma_f32_16x16x4_f32(false, a1, false, b1, (short)0, c, false, false);
  }

  const int n = tile_n * 16 + lo;
  const float addn = bias[n];
  float wcn = 0.f, bcn = 0.f;
  if (wc) { wcn = wc[n]; bcn = bcv[n]; }
#pragma unroll
  for (int j = 0; j < 8; ++j) {
    const int m = tile_m * 16 + hi * 8 + j;
    float v = c[j] + addn;
    if (cov) v += cov[m] * wcn + bcn;
    out[(size_t)m * N + n] = v;
  }
}

// ---------------------------------------------------------------------------
// Fused scores + softmax:
//   score[b,r,s] = Watt . tanh(qc[b,s,:] + rp[r,:]) + batt  -> softmax over s
// One block per b (256 thr = 8 waves).  lane == s (S == 32 == wave32),
// so the softmax is pure cross-lane shuffles.  qc[b] staged in LDS with a
// 516-float row stride -> b128 reads hit every bank exactly twice (optimal).
// ---------------------------------------------------------------------------
#define QSTRIDE 516

__global__ void scores_softmax(const float* __restrict__ qc,
                               const float* __restrict__ rp,
                               const float* __restrict__ Watt,
                               const float* __restrict__ batt,
                               const unsigned char* __restrict__ mask,
                               float* __restrict__ alpha) {
  __shared__ float qcs[S_ * QSTRIDE];
  __shared__ float wa[DR_];

  const int b = blockIdx.x;
  const int t = threadIdx.x;

  // cooperative load of qc[b] (32 x 512 f32) as float4
  for (int i = t; i < S_ * (DR_ / 4); i += 256) {
    const int s  = i >> 7;        // / 128
    const int d4 = i & 127;
    v4f v = *(const v4f*)(qc + ((size_t)(b * S_ + s)) * DR_ + d4 * 4);
    *(v4f*)(qcs + s * QSTRIDE + d4 * 4) = v;
  }
  for (int i = t; i < DR_; i += 256) wa[i] = Watt[i];
  __syncthreads();

  const int wv = t >> 5;
  const int s  = t & 31;
  const float batt0  = batt[0];
  const bool  masked = mask[b * S_ + s] != 0;
  const float* qrow  = qcs + s * QSTRIDE;

  for (int r = wv; r < R_; r += 8) {
    const float* rrow = rp + (size_t)r * DR_;
    float acc = 0.f;
#pragma unroll 8
    for (int d4 = 0; d4 < DR_ / 4; ++d4) {
      v4f rv = *(const v4f*)(rrow + d4 * 4);   // uniform across lanes (L2-hot)
      v4f qv = *(const v4f*)(qrow + d4 * 4);   // LDS, conflict-free
      v4f wv4 = *(const v4f*)(wa + d4 * 4);    // LDS broadcast
      acc += wv4[0] * fast_tanh(qv[0] + rv[0]);
      acc += wv4[1] * fast_tanh(qv[1] + rv[1]);
      acc += wv4[2] * fast_tanh(qv[2] + rv[2]);
      acc += wv4[3] * fast_tanh(qv[3] + rv[3]);
    }
    float score = masked ? -INFINITY : (acc + batt0);

    // wave32 softmax across lanes (lane == s)
    float mx = score;
    for (int off = 16; off > 0; off >>= 1)
      mx = fmaxf(mx, __shfl_xor(mx, off, 32));
    float e = __expf(score - mx);
    float sm = e;
    for (int off = 16; off > 0; off >>= 1)
      sm += __shfl_xor(sm, off, 32);

    alpha[((size_t)(b * R_ + r)) * S_ + s] = e / sm;
  }
}

// ---------------------------------------------------------------------------
// Gathered context: only the A=64 selected rows per batch are computed.
//   out[b,a,w] = sum_s alpha[b, r_space[b,a], s] * q[b,s,w]
// One block per (b,a); 128 threads, 4 outputs each; q tile is L2-resident.
// ---------------------------------------------------------------------------
__global__ void ctx_gather(const float* __restrict__ q,
                           const float* __restrict__ alpha,
                           const int* __restrict__ r_space,
                           float* __restrict__ out_ctx,
                           float* __restrict__ out_alpha) {
  const int ba = blockIdx.x;           // = b*A_ + a
  const int b  = ba >> 6;
  const int t  = threadIdx.x;

  __shared__ float al[S_];
  const int rr = r_space[ba];
  if (t < S_) {
    float v = alpha[((size_t)(b * R_ + rr)) * S_ + t];
    al[t] = v;
    out_alpha[(size_t)ba * S_ + t] = v;
  }
  __syncthreads();

  const float* qb = q + ((size_t)b * S_) * DW_;
  for (int w = t; w < DW_; w += 128) {
    float acc = 0.f;
#pragma unroll
    for (int s = 0; s < S_; ++s)
      acc = fmaf(al[s], qb[(size_t)s * DW_ + w], acc);
    out_ctx[(size_t)ba * DW_ + w] = acc;
  }
}

// ---------------------------------------------------------------------------
extern "C" void kernel_launch(void* const* d_in, const int* in_sizes, int n_in,
                              void* d_out, int out_size, void* d_ws, size_t ws_size,
                              hipStream_t stream) {
  const float*         qv   = (const float*)d_in[0];          // [B,S,DW]
  const unsigned char* mask = (const unsigned char*)d_in[1];  // [B,S] bool
  const int*           rsp  = (const int*)d_in[2];            // [B,A] i32
  const float*         cov  = (const float*)d_in[3];          // [B,S]
  const float*         rel  = (const float*)d_in[4];          // [R,DR]
  const float*         Wq   = (const float*)d_in[5];          // [DR,DW]
  const float*         bq   = (const float*)d_in[6];          // [DR]
  const float*         Wr   = (const float*)d_in[7];          // [DR,DR]
  const float*         br   = (const float*)d_in[8];          // [DR]
  const float*         Wc   = (const float*)d_in[9];          // [DR,1]
  const float*         bc   = (const float*)d_in[10];         // [DR]
  const float*         Watt = (const float*)d_in[11];         // [1,DR]
  const float*         batt = (const float*)d_in[12];         // [1]

  float* outp      = (float*)d_out;
  float* out_ctx   = outp;                               // [B,A,DW]
  float* out_alpha = outp + (size_t)B_ * A_ * DW_;       // [B,A,S]

  float* ws    = (float*)d_ws;
  float* rp    = ws;                                     // R*DR      (256 KB)
  float* qc    = rp + (size_t)R_ * DR_;                  // B*S*DR    (4 MB)
  float* alpha = qc + (size_t)B_ * S_ * DR_;             // B*R*S     (1 MB)

  // rp = rel_emb @ Wr^T + br
  gemm_nt_wmma<<<dim3(R_ / 64, DR_ / 16), 128, 0, stream>>>(
      rel, Wr, rp, R_, DR_, DR_, br, nullptr, nullptr, nullptr);

  // qc = q @ Wq^T + bq + cov*Wc + bc   (coverage linear fused into epilogue)
  gemm_nt_wmma<<<dim3((B_ * S_) / 64, DR_ / 16), 128, 0, stream>>>(
      qv, Wq, qc, B_ * S_, DR_, DW_, bq, cov, Wc, bc);

  // fused tanh-attention scores + softmax over S
  scores_softmax<<<B_, 256, 0, stream>>>(qc, rp, Watt, batt, mask, alpha);

  // gathered context + selected alphas
  ctx_gather<<<B_ * A_, 128, 0, stream>>>(qv, alpha, rsp, out_ctx, out_alpha);
}